// MihGNNEmbeddingTest3_56633438765545
// MI455X (gfx1250) — compile-verified
//
#include <hip/hip_runtime.h>
#include <hip/hip_bf16.h>
#include <stddef.h>

// ---------------------------------------------------------------------------
// MihGNNEmbeddingTest3: algebraically restructured.
//   P = A_s @ emb            (8192x8192 @ 8192x256)  -- f32 WMMA GEMM
//   Weff/beff = fold of the 3 affine layers (no nonlinearity in _mlp!)
//   H = P @ Weff^T + beff    (8192x256 @ 256x256)    -- same f32 WMMA GEMM
//   out[e] = relu(|<H[s],H[d]>| / (||H[s]|| ||H[d]||))
// GEMM: 128x64 block tile, Kc=32, double-buffered LDS, register-stage
// software pipeline, B staged transposed so both A and B fragments are
// single aligned ds_load_b64s.
// ---------------------------------------------------------------------------

typedef __attribute__((ext_vector_type(2))) float v2f;
typedef __attribute__((ext_vector_type(8))) float v8f;

#define NROWS 8192
#define DIM   256
#define NEDGE 8192

#define BM 128
#define BN 64
#define KC 32
#define LDA 34   // even padded stride: conflict-free & keeps b64 alignment
#define LDBT 34  // BsT[n][k] stride, same property

__device__ __forceinline__ v8f wmma_f32(v2f a, v2f b, v8f c) {
  // (neg_a, A, neg_b, B, c_mod, C, reuse_a, reuse_b)
  return __builtin_amdgcn_wmma_f32_16x16x4_f32(false, a, false, b, (short)0, c,
                                               false, false);
}

__device__ __forceinline__ void load_tile(const float* __restrict__ A,
                                          const float* __restrict__ B,
                                          int block_m, int block_n, int k0,
                                          int K, int N, int tid,
                                          float4 ra[4], float4 rb[2]) {
#pragma unroll
  for (int i = 0; i < 4; ++i) {
    const int idx = tid + i * 256;          // 0..1023
    const int row = idx >> 3;               // 0..127
    const int col = (idx & 7) << 2;         // 0..28
    const float* p = A + (size_t)(block_m + row) * K + k0 + col;
    ra[i] = *(const float4*)p;
    if (k0 + KC < K)  // warm cache one K-tile ahead of this load
      __builtin_prefetch(p + KC, 0, 0);
  }
#pragma unroll
  for (int i = 0; i < 2; ++i) {
    const int idx = tid + i * 256;          // 0..511
    const int row = idx >> 4;               // 0..31 (k)
    const int col = (idx & 15) << 2;        // 0..60 (n)
    rb[i] = *(const float4*)(B + (size_t)(k0 + row) * N + block_n + col);
  }
}

__device__ __forceinline__ void store_tile(float* __restrict__ As,
                                           float* __restrict__ BsT, int tid,
                                           const float4 ra[4],
                                           const float4 rb[2]) {
#pragma unroll
  for (int i = 0; i < 4; ++i) {
    const int idx = tid + i * 256;
    const int row = idx >> 3;
    const int col = (idx & 7) << 2;
    float* d = &As[row * LDA + col];
    d[0] = ra[i].x; d[1] = ra[i].y; d[2] = ra[i].z; d[3] = ra[i].w;
  }
#pragma unroll
  for (int i = 0; i < 2; ++i) {
    const int idx = tid + i * 256;
    const int row = idx >> 4;               // k
    const int col = (idx & 15) << 2;        // n
    BsT[(col + 0) * LDBT + row] = rb[i].x;  // transpose: BsT[n][k]
    BsT[(col + 1) * LDBT + row] = rb[i].y;
    BsT[(col + 2) * LDBT + row] = rb[i].z;
    BsT[(col + 3) * LDBT + row] = rb[i].w;
  }
}

// C[M,N] = A[M,K] @ B[K,N] (+ bias[N]).  M%128==0, N%64==0, K%32==0.
__global__ __launch_bounds__(256) void gemm_f32_wmma(
    const float* __restrict__ A, const float* __restrict__ B,
    const float* __restrict__ bias, float* __restrict__ C,
    int M, int N, int K)
{
  __shared__ float As[2][BM * LDA];
  __shared__ float BsT[2][BN * LDBT];

  const int tid  = threadIdx.x;
  const int lane = tid & 31;
  const int lo   = lane & 15;
  const int hi   = lane >> 4;
  const int wave = tid >> 5;          // 0..7
  const int m_off = (wave & 3) * 32;  // wave macro-tile inside block tile
  const int n_off = (wave >> 2) * 32;

  const int block_m = blockIdx.x * BM;
  const int block_n = blockIdx.y * BN;

  v8f acc[2][2];
#pragma unroll
  for (int i = 0; i < 2; ++i)
#pragma unroll
    for (int j = 0; j < 2; ++j) acc[i][j] = (v8f){};

  float4 ra[4];
  float4 rb[2];

  // prologue: stage tile 0
  load_tile(A, B, block_m, block_n, 0, K, N, tid, ra, rb);
  store_tile(As[0], BsT[0], tid, ra, rb);
  __syncthreads();

  const int nk = K / KC;
  for (int t = 0; t < nk; ++t) {
    const int cur = t & 1;
    const bool has_next = (t + 1) < nk;

    // issue next tile's global loads; they fly under the WMMA block below
    if (has_next)
      load_tile(A, B, block_m, block_n, (t + 1) * KC, K, N, tid, ra, rb);

    const float* __restrict__ Ac = As[cur];
    const float* __restrict__ Bc = BsT[cur];

#pragma unroll
    for (int kk = 0; kk < KC; kk += 4) {
      v2f a0, a1, b0, b1;
      {
        const int base = (m_off + lo) * LDA + kk + 2 * hi;   // even -> b64 ok
        a0[0] = Ac[base];
        a0[1] = Ac[base + 1];
        a1[0] = Ac[base + 16 * LDA];
        a1[1] = Ac[base + 16 * LDA + 1];
      }
      {
        const int base = (n_off + lo) * LDBT + kk + 2 * hi;  // even -> b64 ok
        b0[0] = Bc[base];
        b0[1] = Bc[base + 1];
        b1[0] = Bc[base + 16 * LDBT];
        b1[1] = Bc[base + 16 * LDBT + 1];
      }
      acc[0][0] = wmma_f32(a0, b0, acc[0][0]);
      acc[0][1] = wmma_f32(a0, b1, acc[0][1]);
      acc[1][0] = wmma_f32(a1, b0, acc[1][0]);
      acc[1][1] = wmma_f32(a1, b1, acc[1][1]);
    }

    // stage next tile into the alternate buffer (safe: last read of it
    // completed before the previous barrier), then one barrier per iter
    if (has_next) store_tile(As[cur ^ 1], BsT[cur ^ 1], tid, ra, rb);
    __syncthreads();
  }

  // ---- epilogue: C/D layout -> row = base + r + 8*hi, col = base + lo ----
#pragma unroll
  for (int ni = 0; ni < 2; ++ni) {
    const int col = block_n + n_off + ni * 16 + lo;
    const float bv = bias ? bias[col] : 0.0f;
#pragma unroll
    for (int mi = 0; mi < 2; ++mi) {
      const int row0 = block_m + m_off + mi * 16 + hi * 8;
#pragma unroll
      for (int r = 0; r < 8; ++r) {
        C[(size_t)(row0 + r) * N + col] = acc[mi][ni][r] + bv;
      }
    }
  }
}

// W10 = W1 @ W0 ; b1eff = W1 @ b0 + b1
__global__ __launch_bounds__(256) void fold1(const float* __restrict__ Ws,
                                             const float* __restrict__ bs,
                                             float* __restrict__ W10,
                                             float* __restrict__ b1eff)
{
  const float* W0 = Ws;
  const float* W1 = Ws + DIM * DIM;
  const int i = blockIdx.x;
  const int j = threadIdx.x;
  float acc = 0.0f;
  for (int k = 0; k < DIM; ++k) acc += W1[i * DIM + k] * W0[k * DIM + j];
  W10[i * DIM + j] = acc;
  if (j == 0) {
    float b = 0.0f;
    for (int k = 0; k < DIM; ++k) b += W1[i * DIM + k] * bs[k];
    b1eff[i] = b + bs[DIM + i];
  }
}

// WeffT[k][n] = (W2 @ W10)[n][k] ; beff = W2 @ b1eff + b2
__global__ __launch_bounds__(256) void fold2(const float* __restrict__ Ws,
                                             const float* __restrict__ bs,
                                             const float* __restrict__ W10,
                                             const float* __restrict__ b1eff,
                                             float* __restrict__ WeffT,
                                             float* __restrict__ beff)
{
  const float* W2 = Ws + 2 * DIM * DIM;
  const int n = blockIdx.x;
  const int k = threadIdx.x;
  float acc = 0.0f;
  for (int t = 0; t < DIM; ++t) acc += W2[n * DIM + t] * W10[t * DIM + k];
  WeffT[k * DIM + n] = acc;
  if (k == 0) {
    float b = 0.0f;
    for (int t = 0; t < DIM; ++t) b += W2[n * DIM + t] * b1eff[t];
    beff[n] = b + bs[2 * DIM + n];
  }
}

// One wave32 per edge: cosine score with float4 loads + shfl_xor reductions.
__global__ __launch_bounds__(256) void edge_score(const int* __restrict__ edges,
                                                  const float* __restrict__ H,
                                                  float* __restrict__ out)
{
  const int warp = (blockIdx.x * blockDim.x + threadIdx.x) >> 5;
  const int lane = threadIdx.x & 31;
  if (warp >= NEDGE) return;
  const int s = edges[warp * 2 + 0];
  const int d = edges[warp * 2 + 1];
  const float4* hs = (const float4*)(H + (size_t)s * DIM);
  const float4* hd = (const float4*)(H + (size_t)d * DIM);
  float dot = 0.0f, ns = 0.0f, nd = 0.0f;
#pragma unroll
  for (int i = lane; i < DIM / 4; i += 32) {  // 2 iterations
    const float4 a = hs[i];
    const float4 b = hd[i];
    dot += a.x * b.x + a.y * b.y + a.z * b.z + a.w * b.w;
    ns  += a.x * a.x + a.y * a.y + a.z * a.z + a.w * a.w;
    nd  += b.x * b.x + b.y * b.y + b.z * b.z + b.w * b.w;
  }
#pragma unroll
  for (int off = 16; off; off >>= 1) {
    dot += __shfl_xor(dot, off, 32);
    ns  += __shfl_xor(ns,  off, 32);
    nd  += __shfl_xor(nd,  off, 32);
  }
  if (lane == 0) {
    const float c = fabsf(dot) / (sqrtf(ns) * sqrtf(nd));
    out[warp] = fmaxf(c, 0.0f);
  }
}

extern "C" void kernel_launch(void* const* d_in, const int* in_sizes, int n_in,
                              void* d_out, int out_size, void* d_ws, size_t ws_size,
                              hipStream_t stream) {
  const int*   edges = (const int*)d_in[0];
  const float* A_s   = (const float*)d_in[1];
  const float* emb   = (const float*)d_in[2];
  const float* Ws    = (const float*)d_in[3];
  const float* bs    = (const float*)d_in[4];
  float* out = (float*)d_out;

  // workspace layout (floats): P(8192x256) | H(8192x256) | W10 | WeffT | b1eff | beff
  float* P     = (float*)d_ws;
  float* H     = P + (size_t)NROWS * DIM;
  float* W10   = H + (size_t)NROWS * DIM;
  float* WeffT = W10 + DIM * DIM;
  float* b1eff = WeffT + DIM * DIM;
  float* beff  = b1eff + DIM;

  fold1<<<DIM, DIM, 0, stream>>>(Ws, bs, W10, b1eff);
  fold2<<<DIM, DIM, 0, stream>>>(Ws, bs, W10, b1eff, WeffT, beff);

  // P = A_s @ emb   (M=8192, N=256, K=8192)
  gemm_f32_wmma<<<dim3(NROWS / BM, DIM / BN), 256, 0, stream>>>(
      A_s, emb, nullptr, P, NROWS, DIM, NROWS);

  // H = P @ WeffT + beff   (M=8192, N=256, K=256)
  gemm_f32_wmma<<<dim3(NROWS / BM, DIM / BN), 256, 0, stream>>>(
      P, WeffT, beff, H, NROWS, DIM, DIM);

  edge_score<<<NEDGE / 8, 256, 0, stream>>>(edges, H, out);
}